// CropAndResize_16312285790572
// MI455X (gfx1250) — compile-verified
//
#include <hip/hip_runtime.h>
#include <hip/hip_bf16.h>

// Problem constants (from reference setup_inputs / _crop_and_resize)
#define IMG_B   16
#define IMG_C   64
#define IMG_H   512
#define IMG_W   512
#define NBOX    512
#define CROP_H  28
#define CROP_W  28
#define NPIX    (CROP_H * CROP_W)   // 784
#define CPB     8                   // channels per block (32 loads in flight/thread)
#define CGROUPS (IMG_C / CPB)       // 8

__global__ __launch_bounds__(256)
void CropAndResize_16312285790572_kernel(const float* __restrict__ image,
                                         const float* __restrict__ boxes,
                                         const int*   __restrict__ box_ind,
                                         float* __restrict__ out)
{
    // Per-box tap tables, shared across all channels handled by this block.
    __shared__ int   s_xlf[CROP_W], s_xrt[CROP_W], s_xok[CROP_W];
    __shared__ float s_lx [CROP_W];
    __shared__ int   s_yt [CROP_H], s_yb [CROP_H], s_yok[CROP_H];
    __shared__ float s_ly [CROP_H];

    const int box = (int)blockIdx.x >> 3;         // 512 boxes
    const int c0  = ((int)blockIdx.x & 7) * CPB;  // channel group base
    const int tid = (int)threadIdx.x;

    // Uniform (blockIdx-derived) -> scalar loads
    const float y1 = boxes[box * 4 + 0];
    const float x1 = boxes[box * 4 + 1];
    const float y2 = boxes[box * 4 + 2];
    const float x2 = boxes[box * 4 + 3];
    const int   b  = box_ind[box];

    if (tid < CROP_W) {
        // x-tap table (exact order of ops as reference: *(W-1) then /(CROP_W-1))
        const float wsc  = (x2 - x1) * 511.0f / 27.0f;
        const float in_x = x1 * 511.0f + (float)tid * wsc;
        const float fx   = floorf(in_x);
        s_xok[tid] = (in_x >= 0.0f) && (in_x <= 511.0f);
        s_xlf[tid] = min(max((int)fx, 0), IMG_W - 1);
        s_xrt[tid] = min(max((int)ceilf(in_x), 0), IMG_W - 1);
        s_lx [tid] = in_x - fx;
    } else if (tid < CROP_W + CROP_H) {
        // y-tap table
        const int   iy   = tid - CROP_W;
        const float hsc  = (y2 - y1) * 511.0f / 27.0f;
        const float in_y = y1 * 511.0f + (float)iy * hsc;
        const float fy   = floorf(in_y);
        s_yok[iy] = (in_y >= 0.0f) && (in_y <= 511.0f);
        s_yt [iy] = min(max((int)fy, 0), IMG_H - 1);
        s_yb [iy] = min(max((int)ceilf(in_y), 0), IMG_H - 1);
        s_ly [iy] = in_y - fy;
    }
    __syncthreads();

    const unsigned cstride = (unsigned)(IMG_H * IMG_W);            // 1<<18
    const float* __restrict__ imgb = image + ((size_t)(b * IMG_C + c0) << 18);
    float* __restrict__ outb = out + (size_t)(box * IMG_C + c0) * NPIX;

    // 784 pixels / 256 threads: p = tid, tid+256, tid+512, tid+768(partial).
    // Adjacent lanes -> adjacent x taps in the same image row -> line-coalesced
    // gathers; p is the fast output axis -> fully coalesced 32-wide stores.
    for (int p = tid; p < NPIX; p += 256) {
        const int y = p / CROP_W;
        const int x = p - y * CROP_W;

        const int   yt = s_yt[y], yb = s_yb[y];
        const int   xl = s_xlf[x], xr = s_xrt[x];
        const float ly = s_ly[y], lx = s_lx[x];
        const bool  ok = (s_yok[y] != 0) && (s_xok[x] != 0);

        const unsigned oTL = ((unsigned)yt << 9) + (unsigned)xl;
        const unsigned oTR = ((unsigned)yt << 9) + (unsigned)xr;
        const unsigned oBL = ((unsigned)yb << 9) + (unsigned)xl;
        const unsigned oBR = ((unsigned)yb << 9) + (unsigned)xr;

        // Issue all 32 tap loads for the 8 channels up front (indices always
        // clamped in-bounds so loads are unconditionally safe; mask applied at
        // the end exactly like jnp.where). Maximizes loads-in-flight per wave.
        float tl[CPB], tr[CPB], bl[CPB], br[CPB];
        #pragma unroll
        for (int c = 0; c < CPB; ++c) {
            const float* __restrict__ g = imgb + (unsigned)c * cstride;
            tl[c] = g[oTL];
            tr[c] = g[oTR];
            bl[c] = g[oBL];
            br[c] = g[oBR];
        }

        // CDNA5 global_prefetch_b8: pre-touch next pixel-group's source rows
        // while this group's gathers are in flight.
        const int pn = p + 256;
        if (pn < NPIX) {
            const int yn = pn / CROP_W;
            const int xn = pn - yn * CROP_W;
            const unsigned nTL = ((unsigned)s_yt[yn] << 9) + (unsigned)s_xlf[xn];
            const unsigned nBL = ((unsigned)s_yb[yn] << 9) + (unsigned)s_xlf[xn];
            __builtin_prefetch(imgb + nTL, 0, 0);
            __builtin_prefetch(imgb + nBL, 0, 0);
        }

        #pragma unroll
        for (int c = 0; c < CPB; ++c) {
            const float top = tl[c] + (tr[c] - tl[c]) * lx;
            const float bot = bl[c] + (br[c] - bl[c]) * lx;
            const float v   = top + (bot - top) * ly;
            outb[(unsigned)c * NPIX + (unsigned)p] = ok ? v : 0.0f;
        }
    }
}

extern "C" void kernel_launch(void* const* d_in, const int* in_sizes, int n_in,
                              void* d_out, int out_size, void* d_ws, size_t ws_size,
                              hipStream_t stream) {
    (void)in_sizes; (void)n_in; (void)d_ws; (void)ws_size; (void)out_size;
    const float* image   = (const float*)d_in[0];   // (16, 64, 512, 512) f32
    const float* boxes   = (const float*)d_in[1];   // (512, 4) f32
    const int*   box_ind = (const int*)d_in[2];     // (512,) i32
    float*       out     = (float*)d_out;           // (512, 64, 28, 28) f32

    dim3 grid(NBOX * CGROUPS);  // 4096 blocks
    dim3 block(256);            // 8 wave32s
    CropAndResize_16312285790572_kernel<<<grid, block, 0, stream>>>(image, boxes, box_ind, out);
}